// MultiHashEncoding_79860621902018
// MI455X (gfx1250) — compile-verified
//
#include <hip/hip_runtime.h>

#ifndef __has_builtin
#define __has_builtin(x) 0
#endif

#define THREADS 256
#define LANES_PER_PT 4
#define PTS_PER_BLOCK (THREADS / LANES_PER_PT)   // 64 points per block
#define STAGE_FLOATS (PTS_PER_BLOCK * 3)         // 192 coord floats per block

typedef __attribute__((address_space(1))) int* gint_ptr;
typedef __attribute__((address_space(3))) int* lint_ptr;
typedef __attribute__((ext_vector_type(4))) float v4f;   // native vector for NT stores

__device__ __forceinline__ void fma4(float4& a, float w, const float4& v) {
    a.x = fmaf(w, v.x, a.x);
    a.y = fmaf(w, v.y, a.y);
    a.z = fmaf(w, v.z, a.z);
    a.w = fmaf(w, v.w, a.w);
}

// One grid level: trilinear gather-interp of 4 dims (lane's slice) of the
// 16-dim embedding. Matches reference: corner = clip(floor(c)+b, 0, g-1),
// weight = prod(1 - |corner - c|).
__device__ __forceinline__ void grid_level(
    const float* __restrict__ emb,
    const int H, const int W,                 // dims 1,2 (dim0 only in bounds)
    const float m0, const float m1, const float m2,  // g-1 per dim
    const float c0, const float c1, const float c2,  // scaled coords
    const int sub, float4& acc)
{
    const float f0 = floorf(c0), f1 = floorf(c1), f2 = floorf(c2);

    const float a0 = fminf(fmaxf(f0,       0.0f), m0);
    const float b0 = fminf(fmaxf(f0 + 1.f, 0.0f), m0);
    const float a1 = fminf(fmaxf(f1,       0.0f), m1);
    const float b1 = fminf(fmaxf(f1 + 1.f, 0.0f), m1);
    const float a2 = fminf(fmaxf(f2,       0.0f), m2);
    const float b2 = fminf(fmaxf(f2 + 1.f, 0.0f), m2);

    const float wa0 = 1.0f - fabsf(a0 - c0), wb0 = 1.0f - fabsf(b0 - c0);
    const float wa1 = 1.0f - fabsf(a1 - c1), wb1 = 1.0f - fabsf(b1 - c1);
    const float wa2 = 1.0f - fabsf(a2 - c2), wb2 = 1.0f - fabsf(b2 - c2);

    const int ia0 = (int)a0, ib0 = (int)b0;
    const int ia1 = (int)a1, ib1 = (int)b1;
    const int ia2 = (int)a2, ib2 = (int)b2;

    const int rA = ia0 * H, rB = ib0 * H;
    const int pAA = (rA + ia1) * W, pAB = (rA + ib1) * W;
    const int pBA = (rB + ia1) * W, pBB = (rB + ib1) * W;

    // Lane's 4-dim slice of each 16-float (64B) embedding row.
    const float* base = emb + sub * 4;

    // Issue all 8 gathers up-front -> deep LOADcnt to hide L2 latency.
    const float4 v000 = *(const float4*)(base + (size_t)(pAA + ia2) * 16);
    const float4 v001 = *(const float4*)(base + (size_t)(pAA + ib2) * 16);
    const float4 v010 = *(const float4*)(base + (size_t)(pAB + ia2) * 16);
    const float4 v011 = *(const float4*)(base + (size_t)(pAB + ib2) * 16);
    const float4 v100 = *(const float4*)(base + (size_t)(pBA + ia2) * 16);
    const float4 v101 = *(const float4*)(base + (size_t)(pBA + ib2) * 16);
    const float4 v110 = *(const float4*)(base + (size_t)(pBB + ia2) * 16);
    const float4 v111 = *(const float4*)(base + (size_t)(pBB + ib2) * 16);

    const float wAA = wa0 * wa1, wAB = wa0 * wb1;
    const float wBA = wb0 * wa1, wBB = wb0 * wb1;

    fma4(acc, wAA * wa2, v000);
    fma4(acc, wAA * wb2, v001);
    fma4(acc, wAB * wa2, v010);
    fma4(acc, wAB * wb2, v011);
    fma4(acc, wBA * wa2, v100);
    fma4(acc, wBA * wb2, v101);
    fma4(acc, wBB * wa2, v110);
    fma4(acc, wBB * wb2, v111);
}

__global__ __launch_bounds__(THREADS)
void MultiHashEncoding_79860621902018_kernel(
    const float* __restrict__ inputs,   // [N,3]
    const float* __restrict__ emb0,     // [16,136,241,16]
    const float* __restrict__ emb1,     // [9,69,121,16]
    float* __restrict__ out,            // [N,32]
    const int npts)
{
    __shared__ float s_in[STAGE_FLOATS];

    const int tid = threadIdx.x;
    const size_t blk_base = (size_t)blockIdx.x * STAGE_FLOATS;
    const float* gsrc = inputs + blk_base;

    // --- Stage this block's 192 coord floats into LDS, coalesced, via the
    // --- gfx1250 async global->LDS path (ASYNCcnt).
#if __has_builtin(__builtin_amdgcn_global_load_async_to_lds_b32)
    if (tid < STAGE_FLOATS && (blk_base + tid) < (size_t)npts * 3) {
        __builtin_amdgcn_global_load_async_to_lds_b32(
            (gint_ptr)(void*)(gsrc + tid),
            (lint_ptr)(int*)(&s_in[tid]),
            0, 0);
    }
#if __has_builtin(__builtin_amdgcn_s_wait_asynccnt)
    __builtin_amdgcn_s_wait_asynccnt(0);
#else
    asm volatile("s_wait_asynccnt 0" ::: "memory");
#endif
    __syncthreads();
#else
    if (tid < STAGE_FLOATS && (blk_base + tid) < (size_t)npts * 3) {
        s_in[tid] = gsrc[tid];
    }
    __syncthreads();
#endif

    const int p   = tid >> 2;        // point within block
    const int sub = tid & 3;         // 4-dim slice owner
    const int n   = blockIdx.x * PTS_PER_BLOCK + p;
    if (n >= npts) return;

    const float u0 = s_in[p * 3 + 0];
    const float u1 = s_in[p * 3 + 1];
    const float u2 = s_in[p * 3 + 2];

    float4 acc0 = {0.f, 0.f, 0.f, 0.f};
    float4 acc1 = {0.f, 0.f, 0.f, 0.f};

    // Level 0: grid [16,136,241], scale = g-1 = [15,135,240]
    grid_level(emb0, 136, 241, 15.f, 135.f, 240.f,
               u0 * 15.f, u1 * 135.f, u2 * 240.f, sub, acc0);
    // Level 1: grid [9,69,121], scale = [8,68,120]
    grid_level(emb1, 69, 121, 8.f, 68.f, 120.f,
               u0 * 8.f, u1 * 68.f, u2 * 120.f, sub, acc1);

    // out[n, 0:16] = level0, out[n, 16:32] = level1; each lane stores its
    // 4-dim slice for both levels -> 128B fully contiguous per point.
    // Non-temporal: pure 128MB write stream must not evict the ~38MB of
    // embedding tables from the 192MB L2 that all gathers hit.
    float* op = out + (size_t)n * 32 + sub * 4;
    const v4f nt0 = {acc0.x, acc0.y, acc0.z, acc0.w};
    const v4f nt1 = {acc1.x, acc1.y, acc1.z, acc1.w};
#if __has_builtin(__builtin_nontemporal_store)
    __builtin_nontemporal_store(nt0, (v4f*)(op));
    __builtin_nontemporal_store(nt1, (v4f*)(op + 16));
#else
    *(v4f*)(op)      = nt0;
    *(v4f*)(op + 16) = nt1;
#endif
}

extern "C" void kernel_launch(void* const* d_in, const int* in_sizes, int n_in,
                              void* d_out, int out_size, void* d_ws, size_t ws_size,
                              hipStream_t stream) {
    const float* inputs = (const float*)d_in[0];
    const float* emb0   = (const float*)d_in[1];
    const float* emb1   = (const float*)d_in[2];
    float* out          = (float*)d_out;

    const int npts   = in_sizes[0] / 3;
    const int blocks = (npts + PTS_PER_BLOCK - 1) / PTS_PER_BLOCK;

    hipLaunchKernelGGL(MultiHashEncoding_79860621902018_kernel,
                       dim3(blocks), dim3(THREADS), 0, stream,
                       inputs, emb0, emb1, out, npts);
}